// HGate_549755814192
// MI455X (gfx1250) — compile-verified
//
#include <hip/hip_runtime.h>
#include <hip/hip_bf16.h>

typedef __attribute__((ext_vector_type(2))) float v2f;
typedef __attribute__((ext_vector_type(8))) float v8f;

// Apply W = H (x) H (x) H (8x8) along the top-3-bit axis of the 8192-row state,
// batched over 4096 columns:  out[a,r,b] = sum_a' W[a,a'] x[a',r,b].
// One wave handles a 16x16 D tile = two r-rows (r0,r1) x 16 batch cols, using
// W2 = I2 (x) W (16x16 block-diagonal) and 4 chained V_WMMA_F32_16X16X4_F32.
// Each wave loops over 4 consecutive b-tiles (64 columns).
// All global offsets are 32-bit (array is 33.5M elements) so loads/stores use
// the SADDR + voffset form; non-temporal hints keep the 268MB stream out of L2.
__global__ __launch_bounds__(256) void hgate3_wmma_kernel(
    const float* __restrict__ x, const float* __restrict__ M,
    float* __restrict__ out) {
  constexpr unsigned B  = 4096u;   // batch columns
  constexpr unsigned RB = 1024u;   // rows per a-group (2^10)

  const unsigned lane = threadIdx.x & 31u;
  const unsigned wave = (blockIdx.x << 3) | (threadIdx.x >> 5); // 8 waves/block
  const unsigned bGroup = wave & 63u;   // 64 groups of 64 batch columns
  const unsigned rPair  = wave >> 6;    // 512 pairs of r rows
  const unsigned r0 = rPair << 1;
  const unsigned r1 = r0 + 1u;
  const unsigned b0 = bGroup << 6;

  const unsigned m    = lane & 15u;     // row / column within 16-tile
  const unsigned half = lane >> 4;      // 0: K = 4s+{0,1}, 1: K = 4s+{2,3}

  // 2x2 gate entries (uniform scalar loads). H => m00=m01=m10=+c, m11=-c.
  const float g00 = M[0], g01 = M[1], g10 = M[2], g11 = M[3];

  // A fragments: W2[m][k] = (block match) ? prod over 3 bits of M[bit(m)][bit(k)] : 0
  v2f afrag[4];
#pragma unroll
  for (int s = 0; s < 4; ++s) {
#pragma unroll
    for (int j = 0; j < 2; ++j) {
      const unsigned k = 4u * s + 2u * half + j;
      float v = 0.0f;
      if (((m ^ k) & 8u) == 0u) {  // inside a diagonal 8x8 block
        const unsigned mm = m & 7u, kk = k & 7u;
        v = 1.0f;
#pragma unroll
        for (int q = 0; q < 3; ++q) {
          const unsigned mb = (mm >> q) & 1u, kb = (kk >> q) & 1u;
          v *= mb ? (kb ? g11 : g10) : (kb ? g01 : g00);
        }
      }
      afrag[s][j] = v;
    }
  }

  // 32-bit element offsets for the 8 K-rows this lane supplies to B.
  // K-row k: a = k&7, r = (k<8) ? r0 : r1.
  unsigned rowOff[8];
#pragma unroll
  for (int s = 0; s < 4; ++s) {
#pragma unroll
    for (int j = 0; j < 2; ++j) {
      const unsigned k = 4u * s + 2u * half + j;
      const unsigned a = k & 7u;
      const unsigned r = (k & 8u) ? r1 : r0;
      rowOff[2 * s + j] = (a * RB + r) * B;
    }
  }

  v8f acc[4];
#pragma unroll
  for (int t = 0; t < 4; ++t) {
    const unsigned col = b0 + 16u * t + m;
    v8f d = {};  // C = 0
#pragma unroll
    for (int s = 0; s < 4; ++s) {
      v2f bfrag;
      bfrag[0] = __builtin_nontemporal_load(x + (rowOff[2 * s + 0] + col));
      bfrag[1] = __builtin_nontemporal_load(x + (rowOff[2 * s + 1] + col));
      // 8 args: (neg_a, A, neg_b, B, c_mod, C, reuse_a, reuse_b)
      d = __builtin_amdgcn_wmma_f32_16x16x4_f32(
          false, afrag[s], false, bfrag, (short)0, d, false, false);
    }
    acc[t] = d;
  }

  // D layout: VGPR v -> output row (v + 8*half); half selects r0/r1 group, a = v.
  const unsigned rOut = half ? r1 : r0;
#pragma unroll
  for (int t = 0; t < 4; ++t) {
    const unsigned col = b0 + 16u * t + m;
#pragma unroll
    for (int v = 0; v < 8; ++v) {
      __builtin_nontemporal_store(acc[t][v], out + ((v * RB + rOut) * B + col));
    }
  }
}

extern "C" void kernel_launch(void* const* d_in, const int* in_sizes, int n_in,
                              void* d_out, int out_size, void* d_ws, size_t ws_size,
                              hipStream_t stream) {
  const float* x = (const float*)d_in[0];   // (8192, 4096) fp32
  const float* M = (const float*)d_in[1];   // (2, 2) fp32 Hadamard
  float* out = (float*)d_out;               // (8192, 4096) fp32

  // 512 r-pairs * 64 b-groups = 32768 waves; 8 waves (256 thr) per block.
  dim3 grid(4096), block(256);
  hgate3_wmma_kernel<<<grid, block, 0, stream>>>(x, M, out);
}